// GlobalGCN_43215960933160
// MI455X (gfx1250) — compile-verified
//
#include <hip/hip_runtime.h>

typedef float v2f __attribute__((ext_vector_type(2)));
typedef float v8f __attribute__((ext_vector_type(8)));

#define GCN_D 128          // D_IN == D_OUT == 128
#define GCN_N 100000       // N_NODES, divisible by 16

// ---------------------------------------------------------------------------
// Kernel 1: zero the output accumulator (harness poisons d_out with 0xAA).
// ---------------------------------------------------------------------------
__global__ __launch_bounds__(256) void gcn_zero_kernel(float4* __restrict__ out, int n4) {
    int i = blockIdx.x * blockDim.x + threadIdx.x;
    if (i < n4) {
        float4 z; z.x = 0.f; z.y = 0.f; z.z = 0.f; z.w = 0.f;
        out[i] = z;
    }
}

// ---------------------------------------------------------------------------
// Kernel 2: support = X @ W^T via V_WMMA_F32_16X16X4_F32.
//   grid  = N/16 blocks (one 16-row strip per block)
//   block = 256 threads = 8 waves; wave w owns the 16-col tile at n0 = 16*w
// A-frag (16x4 f32): lanes 0-15 hold K=k,k+1 ; lanes 16-31 hold K=k+2,k+3
// B-frag (4x16 f32): lane n holds B[k..][n] = W[n][k..] (B = W^T)
// C/D (16x16 f32):   lane n in 0-15 -> M=r ; lanes 16-31 -> M=8+r (VGPR r)
// ---------------------------------------------------------------------------
__global__ __launch_bounds__(256) void gcn_gemm_wmma_kernel(const float* __restrict__ X,
                                                            const float* __restrict__ W,
                                                            float* __restrict__ support) {
    const int lane  = threadIdx.x & 31;
    const int wave  = threadIdx.x >> 5;        // 0..7 -> n-tile
    const int m0    = blockIdx.x << 4;         // 16-row strip
    const int n0    = wave << 4;
    const int l15   = lane & 15;
    const int khalf = (lane >> 4) << 1;        // 0 for lanes 0-15, 2 for 16-31

    // Preload all 32 B fragments (W row slice) into registers: 64 VGPRs.
    v2f bfrag[32];
    const float* wrow = W + (size_t)(n0 + l15) * GCN_D;
    #pragma unroll
    for (int kk = 0; kk < 32; ++kk) {
        const float2 t = *(const float2*)(wrow + (kk << 2) + khalf);
        bfrag[kk].x = t.x;
        bfrag[kk].y = t.y;
    }

    v8f acc = {};
    const float* xrow = X + (size_t)(m0 + l15) * GCN_D;
    #pragma unroll
    for (int kk = 0; kk < 32; ++kk) {
        const float2 t = *(const float2*)(xrow + (kk << 2) + khalf);
        v2f a;
        a.x = t.x;
        a.y = t.y;
        // (neg_a, A, neg_b, B, c_mod, C, reuse_a, reuse_b)
        acc = __builtin_amdgcn_wmma_f32_16x16x4_f32(false, a, false, bfrag[kk],
                                                    (short)0, acc, false, false);
    }

    // Store the 16x16 tile.
    const int mlane = m0 + ((lane >> 4) << 3);     // +8 rows for upper half-wave
    float* outp = support + (size_t)mlane * GCN_D + n0 + l15;
    #pragma unroll
    for (int r = 0; r < 8; ++r) {
        outp[(size_t)r * GCN_D] = acc[r];
    }
}

// ---------------------------------------------------------------------------
// Kernel 3: COO SpMM scatter. One wave per edge.
//   lane l gathers float4 l of support[col] (coalesced 512B row, L2-resident),
//   scales by the edge value, and scatter-adds with global_atomic_add_f32.
// Edge metadata loads are wave-uniform -> scalarized by the compiler.
// ---------------------------------------------------------------------------
__global__ __launch_bounds__(256) void gcn_spmm_scatter_kernel(const float* __restrict__ support,
                                                               const int*   __restrict__ rows,
                                                               const int*   __restrict__ cols,
                                                               const float* __restrict__ vals,
                                                               float*       __restrict__ out,
                                                               int n_edges) {
    const int lane = threadIdx.x & 31;
    const int edge = (blockIdx.x * blockDim.x + threadIdx.x) >> 5;
    if (edge >= n_edges) return;

    const int   dst = rows[edge];
    const int   src = cols[edge];
    const float v   = vals[edge];

    const float4 m = ((const float4*)(support + (size_t)src * GCN_D))[lane];
    float* o = out + (size_t)dst * GCN_D + (lane << 2);
    atomicAdd(o + 0, v * m.x);
    atomicAdd(o + 1, v * m.y);
    atomicAdd(o + 2, v * m.z);
    atomicAdd(o + 3, v * m.w);
}

// ---------------------------------------------------------------------------
// Launch: zero out -> WMMA GEMM into d_ws -> edge scatter into d_out.
// ---------------------------------------------------------------------------
extern "C" void kernel_launch(void* const* d_in, const int* in_sizes, int n_in,
                              void* d_out, int out_size, void* d_ws, size_t ws_size,
                              hipStream_t stream) {
    const float* x    = (const float*)d_in[0];   // [N, 128]
    const float* W    = (const float*)d_in[1];   // [128, 128]
    const int*   rows = (const int*)  d_in[2];   // [E]
    const int*   cols = (const int*)  d_in[3];   // [E]
    const float* vals = (const float*)d_in[4];   // [E]
    float*       out  = (float*)d_out;           // [N, 128]
    float*       supp = (float*)d_ws;            // [N, 128] scratch (51.2 MB)

    const int n_nodes = in_sizes[0] / GCN_D;     // 100000
    const int n_edges = in_sizes[2];             // 3200000

    // 1) zero output
    const int n4 = (n_nodes * GCN_D) / 4;
    gcn_zero_kernel<<<(n4 + 255) / 256, 256, 0, stream>>>((float4*)out, n4);

    // 2) support = X @ W^T  (WMMA f32 16x16x4, 8 waves/block = 16x128 strip)
    gcn_gemm_wmma_kernel<<<n_nodes / 16, 256, 0, stream>>>(x, W, supp);

    // 3) out += A_sparse @ support  (one wave per edge, fp32 atomics)
    const long long threads = (long long)n_edges * 32;
    gcn_spmm_scatter_kernel<<<(int)((threads + 255) / 256), 256, 0, stream>>>(
        supp, rows, cols, vals, out, n_edges);
}